// SpKBGATModified_63677185130630
// MI455X (gfx1250) — compile-verified
//
#include <hip/hip_runtime.h>
#include <hip/hip_bf16.h>

// ---------------- problem constants (from reference) ----------------
#define NN      50000     // nodes
#define NEDGE   250000    // base edges
#define NHOPE   125000    // nhop edges
#define ETOT    375000    // total edges
#define DIN_C   100
#define D2_C    200       // HEADS*NHID
#define NREL_C  500
#define BATCH_C 65536

typedef __attribute__((ext_vector_type(2))) float v2f;
typedef __attribute__((ext_vector_type(8))) float v8f;

// ---------------- fp32 WMMA GEMM: C[M x Nc] = A[M x K] * B ----------------
// TRANSB=true : B stored as (Nc, K) row-major, element W[n*ldw + wofs + k]
// TRANSB=false: B stored as (K, Nc) row-major, element W[k*ldw + wofs + n]
// One wave computes a 16x32 C tile (two 16x16 WMMA accumulators sharing the
// A fragment) via V_WMMA_F32_16X16X4_F32. K must be a multiple of 4 (100/200).
// Out-of-range rows/cols are index-CLAMPED (not predicated): a clamped A row /
// B column only affects C rows/cols that the bounded store never writes, so the
// inner loop has zero branches and uses plain b64 loads.
template<bool TRANSB>
__global__ void gemm_wmma_f32(const float* __restrict__ A, int lda,
                              const float* __restrict__ W, int ldw, int wofs,
                              float* __restrict__ C, int ldc, int cofs,
                              int M, int Nc, int K) {
  const int lane = threadIdx.x;          // 0..31 (wave32)
  const int half = lane >> 4;            // 0 = lanes 0-15, 1 = lanes 16-31
  const int l16  = lane & 15;
  const int row0 = (blockIdx.y * blockDim.y + threadIdx.y) * 16;
  const int col0 = blockIdx.x * 32;      // two 16-wide column tiles per wave
  if (row0 >= M) return;                 // uniform per wave

  int ar  = row0 + l16;       if (ar  > M - 1)  ar  = M - 1;   // clamp (see above)
  int bcc0 = col0 + l16;      if (bcc0 > Nc - 1) bcc0 = Nc - 1;
  int bcc1 = col0 + 16 + l16; if (bcc1 > Nc - 1) bcc1 = Nc - 1;

  const float* __restrict__ Arow = A + (size_t)ar * lda;
  const float* __restrict__ W0   = TRANSB ? (W + (size_t)bcc0 * ldw + wofs) : (W + wofs + bcc0);
  const float* __restrict__ W1   = TRANSB ? (W + (size_t)bcc1 * ldw + wofs) : (W + wofs + bcc1);

  v8f acc0 = {};
  v8f acc1 = {};
  #pragma unroll 2
  for (int k = 0; k < K; k += 4) {
    const int k0 = k + 2 * half;         // lanes 0-15: K={k,k+1}; lanes 16-31: K={k+2,k+3}
    const v2f a = *(const v2f*)(Arow + k0);          // global_load_b64, 8B aligned
    v2f b0, b1;
    if (TRANSB) {
      b0 = *(const v2f*)(W0 + k0);
      b1 = *(const v2f*)(W1 + k0);
    } else {
      b0.x = W0[(size_t)(k0) * ldw];
      b0.y = W0[(size_t)(k0 + 1) * ldw];
      b1.x = W1[(size_t)(k0) * ldw];
      b1.y = W1[(size_t)(k0 + 1) * ldw];
    }
    acc0 = __builtin_amdgcn_wmma_f32_16x16x4_f32(false, a, false, b0, (short)0, acc0, false, false);
    acc1 = __builtin_amdgcn_wmma_f32_16x16x4_f32(false, a, false, b1, (short)0, acc1, false, false);
  }

  const int c0 = col0 + l16;
  const int c1 = col0 + 16 + l16;
  #pragma unroll
  for (int v = 0; v < 8; ++v) {          // C: VGPR v -> row (v + 8*half), col = lane's l16
    const int r = row0 + v + 8 * half;
    if (r < M) {
      if (c0 < Nc) C[(size_t)r * ldc + cofs + c0] = acc0[v];
      if (c1 < Nc) C[(size_t)r * ldc + cofs + c1] = acc1[v];
    }
  }
}

// ---------------- row-wise L2 normalize (one wave per row) ----------------
__global__ void l2norm_rows(const float* __restrict__ in, float* __restrict__ out,
                            int rows, int dim) {
  const int w    = (int)((blockIdx.x * blockDim.x + threadIdx.x) >> 5);
  const int lane = threadIdx.x & 31;
  if (w >= rows) return;
  const float* r = in + (size_t)w * dim;
  float v[8];
  float s = 0.0f;
  int c = 0;
  for (int j = lane; j < dim; j += 32, ++c) { v[c] = r[j]; s += v[c] * v[c]; }
  #pragma unroll
  for (int off = 16; off; off >>= 1) s += __shfl_xor(s, off, 32);
  float nrm = sqrtf(s);
  nrm = nrm > 1e-12f ? nrm : 1e-12f;
  const float inv = 1.0f / nrm;
  c = 0;
  for (int j = lane; j < dim; j += 32, ++c) out[(size_t)w * dim + j] = v[c] * inv;
}

// ---------------- fused sparse attention edge pass (one wave per edge) ----------------
// edge_m = srcp[e0] + dstp[e1] + relp[...]; p = dot(edge_m, a2);
// ee = exp(-leaky_relu(p, 0.2)); atomically accumulate ee and ee*edge_m on e0.
__global__ void edge_att(const float* __restrict__ srcp, const float* __restrict__ dstp,
                         const float* __restrict__ relp, const float* __restrict__ a2,
                         const int* __restrict__ edge_list,   // (2, NEDGE)
                         const int* __restrict__ edge_type,   // (NEDGE)
                         const int* __restrict__ nhop,        // (NHOPE, 4)
                         float* __restrict__ hacc, float* __restrict__ rowsum,
                         int dim, int nE, int eTot) {
  const int e    = (int)((blockIdx.x * blockDim.x + threadIdx.x) >> 5);
  const int lane = threadIdx.x & 31;
  if (e >= eTot) return;

  float m[8];
  float p = 0.0f;
  int e0;
  if (e < nE) {
    e0 = edge_list[e];
    const int e1 = edge_list[nE + e];
    const float* __restrict__ r0 = relp + (size_t)edge_type[e] * dim;
    const float* __restrict__ sp = srcp + (size_t)e0 * dim;
    const float* __restrict__ dp = dstp + (size_t)e1 * dim;
    int c = 0;
    for (int j = lane; j < dim; j += 32, ++c) {
      const float v = sp[j] + dp[j] + r0[j];
      m[c] = v;
      p += v * a2[j];
    }
  } else {
    const int i = e - nE;
    e0 = nhop[i * 4 + 3];
    const int e1 = nhop[i * 4 + 0];
    const float* __restrict__ r0 = relp + (size_t)nhop[i * 4 + 1] * dim;
    const float* __restrict__ r1 = relp + (size_t)nhop[i * 4 + 2] * dim;
    const float* __restrict__ sp = srcp + (size_t)e0 * dim;
    const float* __restrict__ dp = dstp + (size_t)e1 * dim;
    int c = 0;
    for (int j = lane; j < dim; j += 32, ++c) {
      const float v = sp[j] + dp[j] + r0[j] + r1[j];
      m[c] = v;
      p += v * a2[j];
    }
  }
  #pragma unroll
  for (int off = 16; off; off >>= 1) p += __shfl_xor(p, off, 32);
  const float lr = p > 0.0f ? p : 0.2f * p;     // leaky_relu
  const float ee = __expf(-lr);                 // exp(-leaky_relu(p))
  if (lane == 0) atomicAdd(&rowsum[e0], ee);
  int c = 0;
  for (int j = lane; j < dim; j += 32, ++c)
    atomicAdd(&hacc[(size_t)e0 * dim + j], ee * m[c]);
}

// ---------------- finalize: divide by rowsum, ELU, write into x slab ----------------
__global__ void finalize_att(const float* __restrict__ hacc, const float* __restrict__ rowsum,
                             float* __restrict__ xout, int dim, int ldo, int cofs) {
  const int n = blockIdx.y;
  const int j = blockIdx.x * blockDim.x + threadIdx.x;
  if (j >= dim) return;
  float d = rowsum[n];
  if (d == 0.0f) d = 1e-12f;
  float v = hacc[(size_t)n * dim + j] / d;
  v = v > 0.0f ? v : (__expf(v) - 1.0f);        // ELU
  xout[(size_t)n * ldo + cofs + j] = v;
}

__global__ void zero_f32(float* __restrict__ p, int n) {
  const int i = blockIdx.x * blockDim.x + threadIdx.x;
  if (i < n) p[i] = 0.0f;
}

__global__ void scatter_mask(const int* __restrict__ bi, float* __restrict__ mask, int b) {
  const int i = blockIdx.x * blockDim.x + threadIdx.x;
  if (i < b) mask[bi[i * 3 + 2]] = 1.0f;        // duplicate writes of same value: benign
}

// out[n] = l2norm( gemm[n] + mask[n]*x2[n] )  (one wave per row, dim=200)
__global__ void final_entity(const float* __restrict__ g, const float* __restrict__ x2,
                             const float* __restrict__ mask, float* __restrict__ out,
                             int rows, int dim) {
  const int w    = (int)((blockIdx.x * blockDim.x + threadIdx.x) >> 5);
  const int lane = threadIdx.x & 31;
  if (w >= rows) return;
  const float mk = mask[w];
  float v[8];
  float s = 0.0f;
  int c = 0;
  for (int j = lane; j < dim; j += 32, ++c) {
    const float t = g[(size_t)w * dim + j] + mk * x2[(size_t)w * dim + j];
    v[c] = t;
    s += t * t;
  }
  #pragma unroll
  for (int off = 16; off; off >>= 1) s += __shfl_xor(s, off, 32);
  float nrm = sqrtf(s);
  nrm = nrm > 1e-12f ? nrm : 1e-12f;
  const float inv = 1.0f / nrm;
  c = 0;
  for (int j = lane; j < dim; j += 32, ++c) out[(size_t)w * dim + j] = v[c] * inv;
}

// ---------------- host side ----------------
static inline void launch_gemm(bool transb, const float* A, int lda,
                               const float* W, int ldw, int wofs,
                               float* C, int ldc, int cofs,
                               int M, int Nc, int K, hipStream_t s) {
  dim3 blk(32, 4);                                       // 4 waves per workgroup
  dim3 grd((Nc + 31) / 32, ((M + 15) / 16 + 3) / 4);     // 16x32 tile per wave
  if (transb)
    gemm_wmma_f32<true><<<grd, blk, 0, s>>>(A, lda, W, ldw, wofs, C, ldc, cofs, M, Nc, K);
  else
    gemm_wmma_f32<false><<<grd, blk, 0, s>>>(A, lda, W, ldw, wofs, C, ldc, cofs, M, Nc, K);
}

extern "C" void kernel_launch(void* const* d_in, const int* in_sizes, int n_in,
                              void* d_out, int out_size, void* d_ws, size_t ws_size,
                              hipStream_t stream) {
  const float* ent      = (const float*)d_in[0];   // (N, 100)
  const float* rel      = (const float*)d_in[1];   // (500, 100)
  /* d_in[2] entity_rank: unused by reference */
  const float* a_heads  = (const float*)d_in[3];   // (2, 100, 300)
  const float* a2_heads = (const float*)d_in[4];   // (2, 100)
  const float* W_gat    = (const float*)d_in[5];   // (100, 200)
  const float* a_out    = (const float*)d_in[6];   // (200, 600)
  const float* a2_out   = (const float*)d_in[7];   // (200,)
  const float* W_ent    = (const float*)d_in[8];   // (100, 200)
  const int* edge_list  = (const int*)d_in[9];     // (2, 250000)
  const int* edge_type  = (const int*)d_in[10];    // (250000,)
  const int* nhop       = (const int*)d_in[11];    // (125000, 4)
  const int* binp       = (const int*)d_in[12];    // (65536, 3)

  float* out     = (float*)d_out;                  // out_entity_1 (N,200)
  float* OUT_REL = out + (size_t)NN * D2_C;        // out_relation_1 (500,200)

  // workspace layout (floats)
  float* f    = (float*)d_ws;
  float* X0   = f;                                 // N*100
  float* X    = f + 5000000;                       // N*200  (layer-1 output)
  float* PS   = f + 15000000;                      // N*200  (src projections)
  float* PD   = f + 25000000;                      // N*200  (dst projections)
  float* HA   = f + 35000000;                      // N*200  (atomic accumulator / ent gemm)
  float* X2   = f + 45000000;                      // N*200  (layer-2 output)
  float* RELP = f + 55000000;                      // 500*200 (relation projections)
  float* RS   = f + 55100000;                      // N (rowsum / mask)

  // 1) x0 = l2norm(entity_embeddings)
  l2norm_rows<<<(NN + 3) / 4, 128, 0, stream>>>(ent, X0, NN, DIN_C);

  // 2) out_relation_1 = rel @ W_gat   (500x200, K=100), straight to output slot
  launch_gemm(false, rel, DIN_C, W_gat, D2_C, 0, OUT_REL, D2_C, 0, NREL_C, D2_C, DIN_C, stream);

  // 3) layer-1 attention, per head
  for (int h = 0; h < 2; ++h) {
    const float* a  = a_heads + (size_t)h * 100 * 300;
    const float* a2 = a2_heads + (size_t)h * 100;
    launch_gemm(true, X0,  DIN_C, a, 300, 0,   PS,   DIN_C, 0, NN,     DIN_C, DIN_C, stream); // x0 @ a_src^T
    launch_gemm(true, X0,  DIN_C, a, 300, 100, PD,   DIN_C, 0, NN,     DIN_C, DIN_C, stream); // x0 @ a_dst^T
    launch_gemm(true, rel, DIN_C, a, 300, 200, RELP, DIN_C, 0, NREL_C, DIN_C, DIN_C, stream); // rel @ a_rel^T
    zero_f32<<<(NN * DIN_C + 255) / 256, 256, 0, stream>>>(HA, NN * DIN_C);
    zero_f32<<<(NN + 255) / 256, 256, 0, stream>>>(RS, NN);
    edge_att<<<(ETOT + 3) / 4, 128, 0, stream>>>(PS, PD, RELP, a2, edge_list, edge_type,
                                                 nhop, HA, RS, DIN_C, NEDGE, ETOT);
    dim3 fg(1, NN);
    finalize_att<<<fg, 128, 0, stream>>>(HA, RS, X, DIN_C, D2_C, h * DIN_C);
  }

  // 4) layer-2 attention (dim 200)
  launch_gemm(true, X,       D2_C, a_out, 600, 0,   PS,   D2_C, 0, NN,     D2_C, D2_C, stream);
  launch_gemm(true, X,       D2_C, a_out, 600, 200, PD,   D2_C, 0, NN,     D2_C, D2_C, stream);
  launch_gemm(true, OUT_REL, D2_C, a_out, 600, 400, RELP, D2_C, 0, NREL_C, D2_C, D2_C, stream);
  zero_f32<<<(NN * D2_C + 255) / 256, 256, 0, stream>>>(HA, NN * D2_C);
  zero_f32<<<(NN + 255) / 256, 256, 0, stream>>>(RS, NN);
  edge_att<<<(ETOT + 3) / 4, 128, 0, stream>>>(PS, PD, RELP, a2_out, edge_list, edge_type,
                                               nhop, HA, RS, D2_C, NEDGE, ETOT);
  dim3 fg2(2, NN);
  finalize_att<<<fg2, 128, 0, stream>>>(HA, RS, X2, D2_C, D2_C, 0);

  // 5) x0 @ W_entities (N x 200, K=100) into HA (reused)
  launch_gemm(false, X0, DIN_C, W_ent, D2_C, 0, HA, D2_C, 0, NN, D2_C, DIN_C, stream);

  // 6) mask from batch_inputs[:,2]
  zero_f32<<<(NN + 255) / 256, 256, 0, stream>>>(RS, NN);
  scatter_mask<<<(BATCH_C + 255) / 256, 256, 0, stream>>>(binp, RS, BATCH_C);

  // 7) out_entity_1 = l2norm(x0@W_ent + mask*x2)
  final_entity<<<(NN + 3) / 4, 128, 0, stream>>>(HA, X2, RS, out, NN, D2_C);
}